// MainModule_3_66297115181533
// MI455X (gfx1250) — compile-verified
//
#include <hip/hip_runtime.h>
#include <math.h>

#define TS    16
#define HALO  18      // TS + 2 halo
#define PSTR  12      // tileM pixel stride in dwords: 16B-aligned AND 12*i mod 64 permutes banks

// 16 connectivity pairs: neighbor index n = J*3 + I  (dy=J-1, dx=I-1),
// mask bit b selects color channel (2+b).
__device__ constexpr int PAIR_N1[16] = {0,6,1,3,1,1,0,2,2,3,0,5,1,1,3,7};
__device__ constexpr int PAIR_M1[16] = {0x5,0x9,0x1,0x4,0x6,0xA,0xA,0x6,0x6,0xA,0xA,0x6,0x5,0x9,0x5,0x9};
__device__ constexpr int PAIR_N2[16] = {8,2,7,5,6,8,7,7,3,8,5,6,5,1,7,5};
__device__ constexpr int PAIR_M2[16] = {0xA,0x6,0x2,0x8,0x9,0x5,0x5,0x9,0x9,0x5,0x5,0x9,0xA,0x6,0xA,0x6};

__device__ __forceinline__ uint32_t lds_byte_off(const void* p) {
    // Flat LDS pointers on gfx1250 carry the LDS byte offset in the low 32 bits.
    return (uint32_t)(uintptr_t)p;
}

template<bool HAVE_PART, bool WRITE_RGB>
__global__ __launch_bounds__(256)
void fuse_level(const float* __restrict__ plot,
                const float* __restrict__ part_in,
                float* __restrict__ out,
                int S,
                const float* __restrict__ pp_ptr)
{
    __shared__ float tileP[HALO * HALO * 7];      // raw plot halo tile: [r][c*7+ch]
    __shared__ float tileM[HALO * HALO * PSTR];   // merged tile, 12-dword pixel stride

    const int tx  = threadIdx.x, ty = threadIdx.y;
    const int tid = ty * TS + tx;
    const int gx0 = blockIdx.x * TS;
    const int gy0 = blockIdx.y * TS;

    // ---------- Stage 1: async global->LDS copy of the plot halo tile ----------
    // 18 rows x 126 dwords, iterated over padded 18 x 128 slots (pure shift/mask indexing).
    const int ROWB = HALO * 7;                     // 126 dwords per halo row
    const uint32_t ldsP = lds_byte_off(tileP);
    const int baseRowPx = (gy0 - 1) * S + (gx0 - 1);   // pixel index of tile origin (may be OOB)
    #pragma unroll
    for (int k = 0; k < 9; ++k) {                  // 9 * 256 = 2304 = 18*128 slots
        const int e  = tid + k * 256;
        const int r  = e >> 7;                     // 0..17
        const int c  = e & 127;                    // 0..127 (valid: <126)
        const int gy = gy0 - 1 + r;
        const int px = gx0 - 1 + c / 7;            // c/7 of a 7-bit value: cheap
        if (c < ROWB && (unsigned)gy < (unsigned)S && (unsigned)px < (unsigned)S) {
            const uint32_t voff = (uint32_t)(((baseRowPx + r * S) * 7 + c) * 4);
            const uint32_t ldst = ldsP + (uint32_t)((r * ROWB + c) * 4);
            // GVS form: mem = SGPR64(base) + VGPR32(offset); dest = LDS byte addr in VGPR.
            asm volatile("global_load_async_to_lds_b32 %0, %1, %2"
                         :: "v"(ldst), "v"(voff), "s"(plot) : "memory");
        }
        // out-of-range plot texels are never read from tileP (stage 2 branches on range)
    }
    asm volatile("s_wait_asynccnt 0x0" ::: "memory");
    __syncthreads();

    // ---------- Stage 2: build merged halo tile in LDS ----------
    const float ppv = HAVE_PART ? pp_ptr[0] : 0.0f;
    #pragma unroll
    for (int k = 0; k < 3; ++k) {                  // 3 * 256 = 768 >= 18*32 padded slots
        const int e = tid + k * 256;
        const int r = e >> 5;                      // 0..17 (e < 576)
        const int c = e & 31;                      // valid: < 18
        if (r < HALO && c < HALO) {
            const int gy = gy0 - 1 + r;
            const int gx = gx0 - 1 + c;
            float m[8];
            if ((unsigned)gy < (unsigned)S && (unsigned)gx < (unsigned)S) {
                const float* q = &tileP[r * ROWB + c * 7];
                const float qw = sqrtf(q[6]);
                if (HAVE_PART) {
                    const int Sh = S >> 1;
                    const float* pcp = part_in + ((size_t)(gy >> 1) * Sh + (gx >> 1)) * 7;
                    const float pw  = pcp[6] * ppv;
                    const float wsm = pw + qw;
                    const float rw  = 1.0f / wsm;
                    #pragma unroll
                    for (int j = 0; j < 6; ++j) m[j] = (pw * pcp[j] + qw * q[j]) * rw;
                    m[6] = log1pf(wsm);
                } else {
                    #pragma unroll
                    for (int j = 0; j < 6; ++j) m[j] = q[j];
                    m[6] = qw;
                }
            } else {
                #pragma unroll
                for (int j = 0; j < 7; ++j) m[j] = 1.0f;   // connectivity border value
            }
            m[7] = 0.0f;
            float4* dm = reinterpret_cast<float4*>(&tileM[(r * HALO + c) * PSTR]);
            dm[0] = make_float4(m[0], m[1], m[2], m[3]);
            dm[1] = make_float4(m[4], m[5], m[6], m[7]);
        }
    }
    __syncthreads();

    // ---------- Stage 3: 16-pair connectivity from LDS ----------
    float nb[9][7];
    #pragma unroll
    for (int J = 0; J < 3; ++J) {
        #pragma unroll
        for (int I = 0; I < 3; ++I) {
            const float4* pm =
                reinterpret_cast<const float4*>(&tileM[((ty + J) * HALO + (tx + I)) * PSTR]);
            const float4 a = pm[0];
            const float4 b = pm[1];
            float* d = nb[J * 3 + I];
            d[0] = a.x; d[1] = a.y; d[2] = a.z; d[3] = a.w;
            d[4] = b.x; d[5] = b.y; d[6] = b.z;
        }
    }

    float acc[7];
    #pragma unroll
    for (int j = 0; j < 7; ++j) acc[j] = nb[4][j];   // results[0]: weight 1, center pixel
    float wsum = 1.0f;

    #pragma unroll
    for (int k = 0; k < 16; ++k) {
        const float* c1 = nb[PAIR_N1[k]];
        const float* c2 = nb[PAIR_N2[k]];
        float s1 = 0.0f, s2 = 0.0f;
        #pragma unroll
        for (int b = 0; b < 4; ++b) {
            if (PAIR_M1[k] & (1 << b)) s1 += c1[2 + b];
            if (PAIR_M2[k] & (1 << b)) s2 += c2[2 + b];
        }
        const float t1 = 2.0f * s1;
        const float t2 = 2.0f * s2;
        const float f1 = (t1 > 0.0f) ? (1.0f + t1) : __expf(t1);  // 1 + elu(t)
        const float f2 = (t2 > 0.0f) ? (1.0f + t2) : __expf(t2);
        const float w1 = c1[6], w2 = c2[6];
        const float wsp = w1 + w2;
        const float f   = f1 * f2;
        const float wf  = wsp * f;
        wsum   += wf;
        acc[6] += wf * wsp;                      // per-pair /ws cancels against *ws
        #pragma unroll
        for (int j = 0; j < 6; ++j) acc[j] += f * (c1[j] * w1 + c2[j] * w2);
    }

    const float rw = 1.0f / wsum;
    const int gy = gy0 + ty, gx = gx0 + tx;
    if (WRITE_RGB) {
        float* o = out + ((size_t)gy * S + gx) * 3;
        o[0] = acc[0] * rw; o[1] = acc[1] * rw; o[2] = acc[2] * rw;
    } else {
        float* o = out + ((size_t)gy * S + gx) * 7;
        #pragma unroll
        for (int j = 0; j < 7; ++j) o[j] = acc[j] * rw;
    }
}

extern "C" void kernel_launch(void* const* d_in, const int* in_sizes, int n_in,
                              void* d_out, int out_size, void* d_ws, size_t ws_size,
                              hipStream_t stream) {
    (void)in_sizes; (void)n_in; (void)out_size; (void)ws_size;
    const float* plot0 = (const float*)d_in[0];   // 1024x1024x7
    const float* plot1 = (const float*)d_in[1];   // 512x512x7
    const float* plot2 = (const float*)d_in[2];   // 256x256x7
    const float* plot3 = (const float*)d_in[3];   // 128x128x7
    const float* pp    = (const float*)d_in[5];   // scalar
    float* outp = (float*)d_out;                  // 1024x1024x3

    float* part3 = (float*)d_ws;                  // 128^2 * 7 floats
    float* part2 = part3 + 128 * 128 * 7;         // 256^2 * 7
    float* part1 = part2 + 256 * 256 * 7;         // 512^2 * 7

    const dim3 blk(TS, TS);
    fuse_level<false, false><<<dim3(128 / TS, 128 / TS), blk, 0, stream>>>(plot3, nullptr, part3, 128, pp);
    fuse_level<true,  false><<<dim3(256 / TS, 256 / TS), blk, 0, stream>>>(plot2, part3, part2, 256, pp);
    fuse_level<true,  false><<<dim3(512 / TS, 512 / TS), blk, 0, stream>>>(plot1, part2, part1, 512, pp);
    fuse_level<true,  true ><<<dim3(1024 / TS, 1024 / TS), blk, 0, stream>>>(plot0, part1, outp, 1024, pp);
}